// CombinedLoss_64226940944624
// MI455X (gfx1250) — compile-verified
//
#include <hip/hip_runtime.h>
#include <hip/hip_bf16.h>

typedef __attribute__((ext_vector_type(2))) float v2f;
typedef __attribute__((ext_vector_type(4))) float v4f;
typedef __attribute__((ext_vector_type(8))) float v8f;
typedef __attribute__((ext_vector_type(4))) int   v4i;

#define HW    (512 * 512)          // 262144 = 2^18
#define TOTF  4194304.0f           // 16 * 512 * 512

// Exact full-wave (32 lane) f32 sum using V_WMMA_F32_16X16X4_F32.
// A holds one partial per lane (other K slot zeroed), B = ones.
// D[m][n] = x_m + x_(m+16) (independent of n). Lanes 0-15 hold D rows 0-7,
// lanes 16-31 rows 8-15 => local row-sum + shfl_xor(16) = total, in all lanes.
__device__ __forceinline__ float wave_reduce32(float x) {
    v2f a; a[0] = x;    a[1] = 0.0f;
    v2f b; b[0] = 1.0f; b[1] = 1.0f;
    v8f c = {};
    c = __builtin_amdgcn_wmma_f32_16x16x4_f32(
            /*neg_a=*/false, a, /*neg_b=*/false, b,
            /*c_mod=*/(short)0, c, /*reuse_a=*/false, /*reuse_b=*/false);
    float s = c[0] + c[1] + c[2] + c[3] + c[4] + c[5] + c[6] + c[7];
    s += __shfl_xor(s, 16, 32);
    return s;
}

// Block reduction of 3 scalars; valid result in thread 0. blockDim.x == 256.
__device__ __forceinline__ void block_reduce3(float& s1, float& s2, float& s3) {
    __shared__ float red[8][3];
    s1 = wave_reduce32(s1);
    s2 = wave_reduce32(s2);
    s3 = wave_reduce32(s3);
    const int lane = threadIdx.x & 31;
    const int wave = threadIdx.x >> 5;
    if (lane == 0) { red[wave][0] = s1; red[wave][1] = s2; red[wave][2] = s3; }
    __syncthreads();
    if (threadIdx.x == 0) {
        float a = 0.f, b = 0.f, c = 0.f;
        const int nw = blockDim.x >> 5;
        for (int w = 0; w < nw; ++w) { a += red[w][0]; b += red[w][1]; c += red[w][2]; }
        s1 = a; s2 = b; s3 = c;
    }
}

// Per-pixel math shared by vector body and scalar tail.
__device__ __forceinline__ void accum_pixel(float p0, float p1, float p2, float p3,
                                            int t, float& s_lp, float& s_un,
                                            float& s_in) {
    // stable log-softmax over the 4 channels
    const float mx  = fmaxf(fmaxf(p0, p1), fmaxf(p2, p3));
    const float se  = expf(p0 - mx) + expf(p1 - mx) + expf(p2 - mx) + expf(p3 - mx);
    const float lse = mx + logf(se);
    const float pt  = (t == 0) ? p0 : (t == 1) ? p1 : (t == 2) ? p2 : p3;
    s_lp += pt - lse;

    // union contribution: preds.sum() + targets.sum()
    const float tf = (float)t;
    s_un += (p0 + p1 + p2 + p3) + tf;

    // faithful buggy intersection: exact float equality vs class index
    s_in += (p0 == tf ? 1.f : 0.f) + (p1 == tf ? 1.f : 0.f)
          + (p2 == tf ? 1.f : 0.f) + (p3 == tf ? 1.f : 0.f);
}

// Pass 1: per-block partials of (sum of target log-probs, union, intersection).
// Each thread handles groups of 4 consecutive pixels -> all-b128 global loads.
__global__ void __launch_bounds__(256)
combined_loss_partials(const float* __restrict__ preds,
                       const int*   __restrict__ tgt,
                       float*       __restrict__ blk,
                       int npix) {
    float s_lp = 0.f, s_un = 0.f, s_in = 0.f;
    const int stride = gridDim.x * blockDim.x;
    const int tid    = blockIdx.x * blockDim.x + threadIdx.x;
    const int ngrp   = npix >> 2;      // HW is a multiple of 4: groups never
                                       // straddle a batch boundary
    for (int q = tid; q < ngrp; q += stride) {
        const int  g    = q << 2;
        const int  n    = g >> 18;                         // pixel -> batch index
        const long base = (long)g + (long)n * (3L * HW);   // n*4*HW + hw
        const v4f c0 = __builtin_nontemporal_load((const v4f*)(preds + base));
        const v4f c1 = __builtin_nontemporal_load((const v4f*)(preds + base + 1L * HW));
        const v4f c2 = __builtin_nontemporal_load((const v4f*)(preds + base + 2L * HW));
        const v4f c3 = __builtin_nontemporal_load((const v4f*)(preds + base + 3L * HW));
        const v4i tv = __builtin_nontemporal_load((const v4i*)(tgt + g));
#pragma unroll
        for (int j = 0; j < 4; ++j)
            accum_pixel(c0[j], c1[j], c2[j], c3[j], tv[j], s_lp, s_un, s_in);
    }
    // scalar tail (empty when npix % 4 == 0)
    for (int g = (ngrp << 2) + tid; g < npix; g += stride) {
        const int  n    = g >> 18;
        const long base = (long)g + (long)n * (3L * HW);
        accum_pixel(preds[base], preds[base + 1L * HW], preds[base + 2L * HW],
                    preds[base + 3L * HW], tgt[g], s_lp, s_un, s_in);
    }
    block_reduce3(s_lp, s_un, s_in);
    if (threadIdx.x == 0) {
        blk[blockIdx.x * 3 + 0] = s_lp;
        blk[blockIdx.x * 3 + 1] = s_un;
        blk[blockIdx.x * 3 + 2] = s_in;
    }
}

// Pass 2: single block folds block partials (fixed order) and emits the loss.
__global__ void __launch_bounds__(256)
combined_loss_finalize(const float* __restrict__ blk, int nb,
                       float* __restrict__ out) {
    float s1 = 0.f, s2 = 0.f, s3 = 0.f;
    for (int i = threadIdx.x; i < nb; i += blockDim.x) {
        s1 += blk[i * 3 + 0];
        s2 += blk[i * 3 + 1];
        s3 += blk[i * 3 + 2];
    }
    block_reduce3(s1, s2, s3);
    if (threadIdx.x == 0) {
        const float loss_ce    = -s1 / TOTF;
        const float uni        = s2;
        const float mean_inter = s3 * (1.0f / 16.0f);
        // mean_i[(2*inter_i + S)/(U + S)] == (2*mean(inter) + S)/(U + S)
        const float dice       = (2.0f * mean_inter + 1e-8f) / (uni + 1e-8f);
        const float loss_dice  = 1.0f - dice;
        out[0] = 0.5f * loss_ce + 0.5f * loss_dice;
    }
}

extern "C" void kernel_launch(void* const* d_in, const int* in_sizes, int n_in,
                              void* d_out, int out_size, void* d_ws, size_t ws_size,
                              hipStream_t stream) {
    const float* preds   = (const float*)d_in[0];   // [16,4,512,512] f32
    const int*   targets = (const int*)d_in[1];     // [16,512,512] int
    float*       out     = (float*)d_out;           // scalar f32
    float*       blk     = (float*)d_ws;            // per-block partials (3 f32 each)

    const int npix = in_sizes[1];                   // 16*512*512

    int nb = 1024;                                  // 12 KB of workspace
    const size_t need = (size_t)nb * 3 * sizeof(float);
    if (ws_size < need) {
        nb = (int)(ws_size / (3 * sizeof(float)));
        if (nb < 1) nb = 1;
    }

    combined_loss_partials<<<nb, 256, 0, stream>>>(preds, targets, blk, npix);
    combined_loss_finalize<<<1, 256, 0, stream>>>(blk, nb, out);
}